// LGCN_Encoder_20083267076914
// MI455X (gfx1250) — compile-verified
//
#include <hip/hip_runtime.h>
#include <cstdint>
#include <cstddef>

// ---------------------------------------------------------------------------
// LightGCN propagation on MI455X (gfx1250).
// acc = ego; x = ego; 3x { x = A@x (COO SpMM, atomic scatter); acc += x }
// out = acc / 4.  Memory-bound (AI ~0.25 F/B) -> optimize data movement only.
// CDNA5 paths used: async DMA global->LDS (NT) for the edge stream,
// s_wait_asynccnt, wave32 16-lane-per-edge float4 gathers (b128),
// global_atomic_add_f32 scatters (resolve in the 192MB L2), global_prefetch.
// ---------------------------------------------------------------------------

#define USER_NUM 500000
#define ITEM_NUM 200000
#define N_NODES  (USER_NUM + ITEM_NUM)
#define NNZ_TOT  22400000
#define EMB      64
#define N_LAYERS 3

#define BLOCK 256          // 8 wave32 per workgroup
#define CHUNK 1024         // edges staged in LDS per block (12 KB of 320 KB)

// ---------------------------------------------------------------------------
// COO SpMM: y[row] += val * x[col].  Defined FIRST so the disasm snippet
// shows the async-DMA staging and the fp32 HW atomics.
// ---------------------------------------------------------------------------
__global__ void __launch_bounds__(BLOCK) spmm_coo(const float* __restrict__ vals,
                                                  const int*   __restrict__ rows,
                                                  const int*   __restrict__ cols,
                                                  const float* __restrict__ x,
                                                  float*       __restrict__ y,
                                                  int nnz) {
  __shared__ __align__(16) float s_val[CHUNK];
  __shared__ __align__(16) int   s_row[CHUNK];
  __shared__ __align__(16) int   s_col[CHUNK];

  const int tid = threadIdx.x;
  const long long base = (long long)blockIdx.x * CHUNK;
  const bool full = (base + CHUNK) <= (long long)nnz;

  if (full) {
    // Each thread stages 4 elements (16 B) per array via one async DMA each.
    // Non-temporal: the edge stream is touched exactly once per layer; keep
    // L2 capacity for the 2x179MB embedding vectors instead.
    const int j = tid * 4;
    uint32_t l_val = (uint32_t)(uintptr_t)(&s_val[j]);  // low 32 bits of a
    uint32_t l_row = (uint32_t)(uintptr_t)(&s_row[j]);  // generic LDS ptr ==
    uint32_t l_col = (uint32_t)(uintptr_t)(&s_col[j]);  // LDS byte offset
    const float* g_val = vals + base + j;
    const int*   g_row = rows + base + j;
    const int*   g_col = cols + base + j;
    asm volatile("global_load_async_to_lds_b128 %0, %1, off th:TH_LOAD_NT"
                 :: "v"(l_val), "v"(g_val) : "memory");
    asm volatile("global_load_async_to_lds_b128 %0, %1, off th:TH_LOAD_NT"
                 :: "v"(l_row), "v"(g_row) : "memory");
    asm volatile("global_load_async_to_lds_b128 %0, %1, off th:TH_LOAD_NT"
                 :: "v"(l_col), "v"(g_col) : "memory");
    asm volatile("s_wait_asynccnt 0x0" ::: "memory");
  } else {
    // tail chunk: guarded scalar staging
    for (int j = tid; j < CHUNK; j += BLOCK) {
      long long e = base + j;
      if (e < (long long)nnz) {
        s_val[j] = vals[e];
        s_row[j] = rows[e];
        s_col[j] = cols[e];
      }
    }
  }
  __syncthreads();

  long long rem = (long long)nnz - base;
  const int nmax = rem < CHUNK ? (int)rem : CHUNK;

  const int d0 = (tid & 15) * 4;        // this lane's 4 embedding dims
  for (int i = tid >> 4; i < nmax; i += BLOCK / 16) {
    // software prefetch of the next edge's gather row (random access)
    int ni = i + BLOCK / 16;
    if (ni < nmax)
      __builtin_prefetch(x + (size_t)s_col[ni] * EMB + d0, 0, 1);

    float v  = s_val[i];
    size_t c = (size_t)s_col[i];
    size_t r = (size_t)s_row[i];

    float4 xv = *(const float4*)(x + c * EMB + d0);   // 16 B coalesced gather
    float* yp = y + r * EMB + d0;
    unsafeAtomicAdd(yp + 0, v * xv.x);                // global_atomic_add_f32
    unsafeAtomicAdd(yp + 1, v * xv.y);
    unsafeAtomicAdd(yp + 2, v * xv.z);
    unsafeAtomicAdd(yp + 3, v * xv.w);
  }
}

// ---------------------------------------------------------------------------
// init: ego = concat(user_emb, item_emb) -> x0 (ws) and acc (out);
// also zeroes the first scatter target (fused, saves a separate sweep).
// ---------------------------------------------------------------------------
__global__ void __launch_bounds__(BLOCK) init_ego(const float* __restrict__ ue,
                                                  const float* __restrict__ ie,
                                                  float* __restrict__ x0,
                                                  float* __restrict__ acc,
                                                  float* __restrict__ zb) {
  const size_t userN4 = (size_t)USER_NUM * (EMB / 4);
  const size_t totN4  = (size_t)N_NODES * (EMB / 4);
  size_t i = (size_t)blockIdx.x * BLOCK + threadIdx.x;   // float4 index
  if (i >= totN4) return;
  float4 v = (i < userN4) ? ((const float4*)ue)[i]
                          : ((const float4*)ie)[i - userN4];
  ((float4*)x0)[i]  = v;
  ((float4*)acc)[i] = v;
  ((float4*)zb)[i]  = make_float4(0.f, 0.f, 0.f, 0.f);
}

// ---------------------------------------------------------------------------
// acc = (acc + x) * s ; optionally zero the dead ping-pong buffer z (it was
// this layer's gather source and becomes next layer's scatter target).
// s = 1 for layers 0..1, 0.25 for the last layer (folds the /4 mean).
// ---------------------------------------------------------------------------
__global__ void __launch_bounds__(BLOCK) add_scale(float* __restrict__ acc,
                                                   const float* __restrict__ x,
                                                   float* __restrict__ z,
                                                   float s, size_t n4) {
  size_t i = (size_t)blockIdx.x * BLOCK + threadIdx.x;
  if (i >= n4) return;
  float4 a = ((float4*)acc)[i];
  float4 b = ((const float4*)x)[i];
  a.x = (a.x + b.x) * s;
  a.y = (a.y + b.y) * s;
  a.z = (a.z + b.z) * s;
  a.w = (a.w + b.w) * s;
  ((float4*)acc)[i] = a;
  if (z) ((float4*)z)[i] = make_float4(0.f, 0.f, 0.f, 0.f);
}

// ---------------------------------------------------------------------------
extern "C" void kernel_launch(void* const* d_in, const int* in_sizes, int n_in,
                              void* d_out, int out_size, void* d_ws, size_t ws_size,
                              hipStream_t stream) {
  const float* ue = (const float*)d_in[0];   // [USER_NUM, 64]
  const float* ie = (const float*)d_in[1];   // [ITEM_NUM, 64]
  const float* av = (const float*)d_in[2];   // [NNZ]
  const int*   ar = (const int*)d_in[3];     // [NNZ]
  const int*   ac = (const int*)d_in[4];     // [NNZ]
  float* acc = (float*)d_out;                // [N_NODES, 64] users then items

  const size_t nfl = (size_t)N_NODES * EMB;
  const size_t n4  = nfl / 4;
  float* bufA = (float*)d_ws;                // ping
  float* bufB = bufA + nfl;                  // pong  (2 x 179.2 MB of ws)

  const int g4 = (int)((n4 + BLOCK - 1) / BLOCK);
  const int gE = (int)(((long long)NNZ_TOT + CHUNK - 1) / CHUNK);

  // ego -> bufA and acc; zero bufB (first scatter target)
  init_ego<<<g4, BLOCK, 0, stream>>>(ue, ie, bufA, acc, bufB);

  float* xin  = bufA;
  float* xout = bufB;
  for (int l = 0; l < N_LAYERS; ++l) {
    spmm_coo<<<gE, BLOCK, 0, stream>>>(av, ar, ac, xin, xout, NNZ_TOT);
    const bool last = (l == N_LAYERS - 1);
    const float s = last ? (1.0f / (N_LAYERS + 1)) : 1.0f;
    // zero xin for reuse as next layer's scatter target (skip on last layer)
    add_scale<<<g4, BLOCK, 0, stream>>>(acc, xout, last ? nullptr : xin, s, n4);
    float* t = xin; xin = xout; xout = t;
  }
}